// GNN_8796093022906
// MI455X (gfx1250) — compile-verified
//
#include <hip/hip_runtime.h>
#include <math.h>

// ---------------------------------------------------------------------------
// 2-layer GCN for MI455X (gfx1250, wave32).
//
//   h1  = relu( D^-1/2 A D^-1/2 (x@W1) + b1 )
//   h2  =       D^-1/2 A D^-1/2 (h1@W2) + b2
//   out = log_softmax(h2)
//
// Normalization folded as: out[d] = dinv[d]*(sum_src hs[src] + hs[d]) + b,
// hs[i] = (x@W)[i]*dinv[i].  Scatter = plain float atomicAdd (L2-resident).
// Layer-1 GEMM uses V_WMMA_F32_16X16X4_F32 (N==HIDDEN==16 tile exact fit).
// ---------------------------------------------------------------------------

typedef __attribute__((ext_vector_type(2))) float v2f;
typedef __attribute__((ext_vector_type(8))) float v8f;

#define NFEAT 128
#define NHID  16

// --- degree = 1 (self loop) ------------------------------------------------
__global__ void k_init_deg(float* __restrict__ deg, int n) {
    int i = blockIdx.x * blockDim.x + threadIdx.x;
    if (i < n) deg[i] = 1.0f;
}

// --- deg[dst] += 1 per edge ------------------------------------------------
__global__ void k_degree(const int* __restrict__ ei, float* __restrict__ deg,
                         int n_edges) {
    int e = blockIdx.x * blockDim.x + threadIdx.x;
    if (e < n_edges) atomicAdd(&deg[ei[n_edges + e]], 1.0f);
}

// --- deg -> rsqrt(deg) in place (deg >= 1 always) --------------------------
__global__ void k_dinv(float* __restrict__ deg, int n) {
    int i = blockIdx.x * blockDim.x + threadIdx.x;
    if (i < n) deg[i] = rsqrtf(deg[i]);
}

// --- layer-1 GEMM: hs1 = (x @ W1) * dinv[row];  agg1 init = hs1 (self loop)
// One wave per 16-row tile. A = 16x4 fp32 frag, B = 4x16 fp32 frag, C 16x16.
__global__ void k_gemm1_wmma(const float* __restrict__ x,
                             const float* __restrict__ W1,
                             const float* __restrict__ dinv,
                             float* __restrict__ hs1,
                             float* __restrict__ agg1,
                             int n_nodes) {
    const int lane = threadIdx.x & 31;
    const int wave = threadIdx.x >> 5;
    const int tile = blockIdx.x * (blockDim.x >> 5) + wave;
    const int base = tile * 16;
    if (base >= n_nodes) return;           // wave-uniform: EXEC stays all-ones

    const int col   = lane & 15;           // M for A-frag, N for B/C frags
    const int khalf = (lane >> 4) << 1;    // lanes 0-15 hold K+0..1, 16-31 K+2..3

    const float* xrow = x  + (size_t)(base + col) * NFEAT + khalf;
    const float* wcol = W1 + (size_t)khalf * NHID + col;   // W1[k][n] row-major

    v8f acc = {0.f, 0.f, 0.f, 0.f, 0.f, 0.f, 0.f, 0.f};
#pragma unroll
    for (int k = 0; k < NFEAT; k += 4) {
        v2f a = *(const v2f*)(xrow + k);                   // x[row][k+khalf, +1]
        v2f b;
        b.x = wcol[k * NHID];                              // W1[k+khalf  ][col]
        b.y = wcol[k * NHID + NHID];                       // W1[k+khalf+1][col]
        acc = __builtin_amdgcn_wmma_f32_16x16x4_f32(
            /*neg_a=*/false, a, /*neg_b=*/false, b,
            /*c_mod=*/(short)0, acc, /*reuse_a=*/false, /*reuse_b=*/false);
    }

    const int mbase = (lane >> 4) << 3;    // upper half-wave holds rows +8
#pragma unroll
    for (int v = 0; v < 8; ++v) {
        const int m  = mbase + v;
        const float s = dinv[base + m];
        const float val = acc[v] * s;
        const size_t off = (size_t)(base + m) * NHID + col;
        hs1[off]  = val;   // message source
        agg1[off] = val;   // self-loop term doubles as accumulator init
    }
}

// --- layer-1 scatter: agg1[dst] += hs1[src], one thread per (edge, float4) -
__global__ void k_scatter1(const int* __restrict__ ei,
                           const float* __restrict__ hs1,
                           float* __restrict__ agg1, int n_edges) {
    int idx = blockIdx.x * blockDim.x + threadIdx.x;
    int e = idx >> 2;
    if (e >= n_edges) return;
    int q   = (idx & 3) << 2;
    int src = ei[e];
    int dst = ei[n_edges + e];
    const float4 v = *(const float4*)(hs1 + (size_t)src * NHID + q);
    float* o = agg1 + (size_t)dst * NHID + q;
    atomicAdd(o + 0, v.x);
    atomicAdd(o + 1, v.y);
    atomicAdd(o + 2, v.z);
    atomicAdd(o + 3, v.w);
}

// --- finalize L1 (+b1, relu) fused with L2 GEMM (16x2) and dinv scaling ----
__global__ void k_mid(const float* __restrict__ agg1,
                      const float* __restrict__ dinv,
                      const float* __restrict__ b1,
                      const float* __restrict__ W2,
                      float* __restrict__ hs2,
                      float* __restrict__ agg2, int n) {
    int i = blockIdx.x * blockDim.x + threadIdx.x;
    if (i >= n) return;
    const float di = dinv[i];
    float z0 = 0.f, z1 = 0.f;
#pragma unroll
    for (int f = 0; f < NHID; ++f) {
        float t = fmaf(di, agg1[(size_t)i * NHID + f], b1[f]);
        t = fmaxf(t, 0.f);                                   // relu
        z0 = fmaf(t, W2[f * 2 + 0], z0);
        z1 = fmaf(t, W2[f * 2 + 1], z1);
    }
    const float o0 = z0 * di, o1 = z1 * di;                  // fold dinv[src]
    hs2[(size_t)i * 2 + 0]  = o0;
    hs2[(size_t)i * 2 + 1]  = o1;
    agg2[(size_t)i * 2 + 0] = o0;                            // self loop init
    agg2[(size_t)i * 2 + 1] = o1;
}

// --- layer-2 scatter: one thread per edge, 2 floats ------------------------
__global__ void k_scatter2(const int* __restrict__ ei,
                           const float* __restrict__ hs2,
                           float* __restrict__ agg2, int n_edges) {
    int e = blockIdx.x * blockDim.x + threadIdx.x;
    if (e >= n_edges) return;
    int src = ei[e];
    int dst = ei[n_edges + e];
    const float2 v = *(const float2*)(hs2 + (size_t)src * 2);
    atomicAdd(agg2 + (size_t)dst * 2 + 0, v.x);
    atomicAdd(agg2 + (size_t)dst * 2 + 1, v.y);
}

// --- finalize L2 (+b2) and 2-class log_softmax -----------------------------
__global__ void k_final(const float* __restrict__ agg2,
                        const float* __restrict__ dinv,
                        const float* __restrict__ b2,
                        float* __restrict__ out, int n) {
    int i = blockIdx.x * blockDim.x + threadIdx.x;
    if (i >= n) return;
    const float di = dinv[i];
    const float z0 = fmaf(di, agg2[(size_t)i * 2 + 0], b2[0]);
    const float z1 = fmaf(di, agg2[(size_t)i * 2 + 1], b2[1]);
    const float m   = fmaxf(z0, z1);
    const float lse = m + __logf(__expf(z0 - m) + __expf(z1 - m));
    out[(size_t)i * 2 + 0] = z0 - lse;
    out[(size_t)i * 2 + 1] = z1 - lse;
}

extern "C" void kernel_launch(void* const* d_in, const int* in_sizes, int n_in,
                              void* d_out, int out_size, void* d_ws, size_t ws_size,
                              hipStream_t stream) {
    const float* x  = (const float*)d_in[0];
    const float* W1 = (const float*)d_in[1];
    const float* b1 = (const float*)d_in[2];
    const float* W2 = (const float*)d_in[3];
    const float* b2 = (const float*)d_in[4];
    const int*   ei = (const int*)d_in[5];

    const int N = in_sizes[0] / NFEAT;   // 100000
    const int E = in_sizes[5] / 2;       // 3200000

    // workspace layout (floats): dinv[N] | hs1[N*16] | agg1[N*16] | hs2[N*2] | agg2[N*2]
    float* ws   = (float*)d_ws;
    float* dinv = ws;
    float* hs1  = dinv + N;
    float* agg1 = hs1 + (size_t)N * NHID;
    float* hs2  = agg1 + (size_t)N * NHID;
    float* agg2 = hs2 + (size_t)N * 2;
    float* out  = (float*)d_out;

    const int T = 256;

    k_init_deg<<<(N + T - 1) / T, T, 0, stream>>>(dinv, N);
    k_degree  <<<(E + T - 1) / T, T, 0, stream>>>(ei, dinv, E);
    k_dinv    <<<(N + T - 1) / T, T, 0, stream>>>(dinv, N);

    const int tiles = N / 16;                       // 6250, N divisible by 16
    const int wavesPerBlock = T / 32;               // 8
    k_gemm1_wmma<<<(tiles + wavesPerBlock - 1) / wavesPerBlock, T, 0, stream>>>(
        x, W1, dinv, hs1, agg1, N);

    k_scatter1<<<((size_t)E * 4 + T - 1) / T, T, 0, stream>>>(ei, hs1, agg1, E);
    k_mid     <<<(N + T - 1) / T, T, 0, stream>>>(agg1, dinv, b1, W2, hs2, agg2, N);
    k_scatter2<<<(E + T - 1) / T, T, 0, stream>>>(ei, hs2, agg2, E);
    k_final   <<<(N + T - 1) / T, T, 0, stream>>>(agg2, dinv, b2, out, N);
}